// TripletMessagePassingLayer_88536455840505
// MI455X (gfx1250) — compile-verified
//
#include <hip/hip_runtime.h>
#include <math.h>

#define NN 50000
#define EE 800000
#define CC 32
#define HH 4
#define STEPS 3
#define NEG_SLOPE 0.2f

typedef __attribute__((ext_vector_type(2))) float v2f;
typedef __attribute__((ext_vector_type(8))) float v8f;

// ---------------------------------------------------------------------------
// WMMA f32 GEMM, fully compile-time shapes:
//   C[M x NCOLS] = A[M x K] * B (+ optional bias & CELU epilogue)
//   BT=0: B is [K x NCOLS] row-major;  BT=1: B is [NCOLS x K] row-major (B^T)
// One wave = one 16-row M tile, NTILES = NCOLS/16 accumulators.
// B staged in LDS pair-interleaved so each fragment is one ds_load_b64:
//   Blds[(p*NCOLS + col)*2 + j] = B[2p + j][col],  p = kk*2 + ksel
// A fragments preloaded as vector b64 global loads before the WMMA chain.
// ---------------------------------------------------------------------------
template <int K, int NCOLS, int BT, int EPI>
__global__ void __launch_bounds__(256) gemm_wmma(
    const float* __restrict__ A, const float* __restrict__ B,
    float* __restrict__ Cmat, const float* __restrict__ bias, int Mtiles)
{
  constexpr int KSTEPS = K / 4;
  constexpr int NTILES = NCOLS / 16;

  __shared__ float Blds[K * NCOLS];

  // Stage B into LDS (pair-interleaved layout)
  for (int idx = threadIdx.x; idx < K * NCOLS; idx += 256) {
    int k   = idx / NCOLS;
    int col = idx - k * NCOLS;
    int p   = k >> 1;
    int j   = k & 1;
    float v = BT ? B[(size_t)col * K + k] : B[(size_t)k * NCOLS + col];
    Blds[(p * NCOLS + col) * 2 + j] = v;
  }
  __syncthreads();

  int wave  = threadIdx.x >> 5;
  int lane  = threadIdx.x & 31;
  int mtile = blockIdx.x * 8 + wave;
  if (mtile >= Mtiles) return;

  int mrow = lane & 15;   // M row within tile (A) / N col within tile (B,C)
  int ksel = lane >> 4;   // which K pair this half-wave holds
  int m0   = mtile * 16;

  // Preload all A fragments: lane needs A[m0+mrow][k0 + 2*ksel .. +1] per kk
  const float* Arow = A + (size_t)(m0 + mrow) * K + 2 * ksel;
  v2f areg[KSTEPS];
  #pragma unroll
  for (int kk = 0; kk < KSTEPS; ++kk)
    areg[kk] = *reinterpret_cast<const v2f*>(Arow + kk * 4);

  v8f acc[NTILES];
  #pragma unroll
  for (int i = 0; i < NTILES; ++i) acc[i] = (v8f)0.f;

  #pragma unroll
  for (int kk = 0; kk < KSTEPS; ++kk) {
    int p = kk * 2 + ksel;
    #pragma unroll
    for (int nt = 0; nt < NTILES; ++nt) {
      int col = nt * 16 + mrow;
      v2f b = *reinterpret_cast<const v2f*>(&Blds[(p * NCOLS + col) * 2]);
      acc[nt] = __builtin_amdgcn_wmma_f32_16x16x4_f32(
          false, areg[kk], false, b, (short)0, acc[nt], false, false);
    }
  }

  // C layout: vgpr r -> row = m0 + r + 8*(lane/16), col = nt*16 + lane%16
  #pragma unroll
  for (int nt = 0; nt < NTILES; ++nt) {
    int col = nt * 16 + mrow;
    float bcol = (EPI == 1) ? bias[col] : 0.f;
    #pragma unroll
    for (int r = 0; r < 8; ++r) {
      int row = m0 + r + 8 * ksel;
      float v = acc[nt][r];
      if (EPI == 1) {
        v += bcol;
        v = (v > 0.f) ? v : (__expf(v) - 1.f);   // CELU(alpha=1)
      }
      Cmat[(size_t)row * NCOLS + col] = v;
    }
  }
}

// ---------------------------------------------------------------------------
// Per-node attention scores: s_i[n,h] = <xn[n,h,:], w_att[h,0:32]>
//                            s_j[n,h] = <xn[n,h,:], w_att[h,64:96]>
// ---------------------------------------------------------------------------
__global__ void scores_kernel(const float* __restrict__ xn,
                              const float* __restrict__ w_att,
                              float* __restrict__ s_i, float* __restrict__ s_j)
{
  int idx = blockIdx.x * blockDim.x + threadIdx.x;
  if (idx >= NN * HH) return;
  int h = idx & 3;
  int n = idx >> 2;
  const float* row = xn + (size_t)n * (HH * CC) + h * CC;
  const float* wi  = w_att + h * (3 * CC);
  const float* wj  = wi + 2 * CC;
  float a = 0.f, b = 0.f;
  #pragma unroll
  for (int c = 0; c < CC; ++c) {
    float xv = row[c];
    a = fmaf(xv, wi[c], a);
    b = fmaf(xv, wj[c], b);
  }
  s_i[idx] = a;
  s_j[idx] = b;
}

// ---------------------------------------------------------------------------
// t[e,h] = edge_attr[e,:] . v[:,h] with v[k,h] = sum_c w_edge[k,h*32+c]*w_e[h,c]
// (w_e = w_att[h, 32:64]).  Constant across steps -> computed once.
// ---------------------------------------------------------------------------
__global__ void t_kernel(const float* __restrict__ edge_attr,
                         const float* __restrict__ w_edge,
                         const float* __restrict__ w_att,
                         float* __restrict__ tbuf)
{
  __shared__ float v[CC * HH];
  if (threadIdx.x < CC * HH) {
    int k = threadIdx.x >> 2;
    int h = threadIdx.x & 3;
    float acc = 0.f;
    #pragma unroll
    for (int c = 0; c < CC; ++c)
      acc = fmaf(w_edge[(size_t)k * (HH * CC) + h * CC + c],
                 w_att[h * (3 * CC) + CC + c], acc);
    v[k * HH + h] = acc;
  }
  __syncthreads();
  for (int e = blockIdx.x * blockDim.x + threadIdx.x; e < EE;
       e += gridDim.x * blockDim.x) {
    const float* ea = edge_attr + (size_t)e * CC;
    float t0 = 0.f, t1 = 0.f, t2 = 0.f, t3 = 0.f;
    #pragma unroll
    for (int k = 0; k < CC; ++k) {
      float ev = ea[k];
      t0 = fmaf(ev, v[k * HH + 0], t0);
      t1 = fmaf(ev, v[k * HH + 1], t1);
      t2 = fmaf(ev, v[k * HH + 2], t2);
      t3 = fmaf(ev, v[k * HH + 3], t3);
    }
    tbuf[(size_t)e * HH + 0] = t0;
    tbuf[(size_t)e * HH + 1] = t1;
    tbuf[(size_t)e * HH + 2] = t2;
    tbuf[(size_t)e * HH + 3] = t3;
  }
}

__global__ void init_amax_kernel(float* __restrict__ amax, int n)
{
  int i = blockIdx.x * blockDim.x + threadIdx.x;
  if (i < n) amax[i] = -INFINITY;
}

__device__ inline void atomicMaxF(float* addr, float val)
{
  unsigned int* p = (unsigned int*)addr;
  unsigned int old = *p;
  while (__uint_as_float(old) < val) {
    unsigned int assumed = old;
    old = atomicCAS(p, assumed, __float_as_uint(val));
    if (old == assumed) break;
  }
}

// Pass 1: logits + leaky-relu + segment max over dst
__global__ void alpha_kernel(const int* __restrict__ src, const int* __restrict__ dst,
                             const float* __restrict__ s_i, const float* __restrict__ s_j,
                             const float* __restrict__ tbuf,
                             float* __restrict__ alpha, float* __restrict__ amax)
{
  int e = blockIdx.x * blockDim.x + threadIdx.x;
  if (e >= EE) return;
  int s = src[e], d = dst[e];
  #pragma unroll
  for (int h = 0; h < HH; ++h) {
    float v = s_i[(size_t)d * HH + h] + tbuf[(size_t)e * HH + h] +
              s_j[(size_t)s * HH + h];
    v = (v > 0.f) ? v : NEG_SLOPE * v;
    alpha[(size_t)e * HH + h] = v;
    atomicMaxF(&amax[(size_t)d * HH + h], v);
  }
}

// Pass 2: ex = exp(alpha - amax[dst]) (in place), denom += ex
__global__ void exp_kernel(const int* __restrict__ dst,
                           const float* __restrict__ amax,
                           float* __restrict__ alpha, float* __restrict__ denom)
{
  int e = blockIdx.x * blockDim.x + threadIdx.x;
  if (e >= EE) return;
  int d = dst[e];
  #pragma unroll
  for (int h = 0; h < HH; ++h) {
    float ex = __expf(alpha[(size_t)e * HH + h] - amax[(size_t)d * HH + h]);
    alpha[(size_t)e * HH + h] = ex;
    atomicAdd(&denom[(size_t)d * HH + h], ex);
  }
}

// Pass 3: recompute ea row on the fly (w_edge cached in LDS, 16KB),
// msg = a * ea * x_j, scatter-add into aggr. 128 threads = one output row.
__global__ void __launch_bounds__(128) msg_aggr_kernel(
    const float* __restrict__ edge_attr, const int* __restrict__ src,
    const int* __restrict__ dst, const float* __restrict__ w_edge,
    const float* __restrict__ xn, const float* __restrict__ ex,
    const float* __restrict__ denom, float* __restrict__ aggr)
{
  __shared__ float wlds[CC * HH * CC];   // 32 x 128
  for (int i = threadIdx.x; i < CC * HH * CC; i += blockDim.x)
    wlds[i] = w_edge[i];
  __syncthreads();

  int hc = threadIdx.x;      // 0..127
  int h  = hc >> 5;
  for (int e = blockIdx.x; e < EE; e += gridDim.x) {
    int s = src[e], d = dst[e];
    const float* ea = edge_attr + (size_t)e * CC;
    float acc = 0.f;
    #pragma unroll
    for (int k = 0; k < CC; ++k)
      acc = fmaf(ea[k], wlds[k * (HH * CC) + hc], acc);
    float a  = ex[(size_t)e * HH + h] / (denom[(size_t)d * HH + h] + 1e-16f);
    float xj = xn[(size_t)s * (HH * CC) + hc];
    atomicAdd(&aggr[(size_t)d * (HH * CC) + hc], a * acc * xj);
  }
}

// ---------------------------------------------------------------------------
// Fused GRU cell + LayerNorm. One wave32 per node; lane = channel (C == 32).
// Cross-lane reductions for mean/var via shfl_xor.
// ---------------------------------------------------------------------------
__global__ void gru_ln_kernel(const float* __restrict__ gi,
                              const float* __restrict__ gh,
                              const float* __restrict__ b_ih,
                              const float* __restrict__ b_hh,
                              float* __restrict__ hbuf,
                              const float* __restrict__ gamma,
                              const float* __restrict__ beta,
                              float* __restrict__ xout)
{
  int gid  = blockIdx.x * blockDim.x + threadIdx.x;
  int node = gid >> 5;
  int lane = gid & 31;
  if (node >= NN) return;

  size_t gbase = (size_t)node * (3 * CC);
  float ir  = gi[gbase + lane]          + b_ih[lane];
  float iz  = gi[gbase + CC + lane]     + b_ih[CC + lane];
  float inn = gi[gbase + 2 * CC + lane] + b_ih[2 * CC + lane];
  float hr  = gh[gbase + lane]          + b_hh[lane];
  float hz  = gh[gbase + CC + lane]     + b_hh[CC + lane];
  float hn  = gh[gbase + 2 * CC + lane] + b_hh[2 * CC + lane];
  float hp  = hbuf[(size_t)node * CC + lane];

  float r = 1.f / (1.f + __expf(-(ir + hr)));
  float z = 1.f / (1.f + __expf(-(iz + hz)));
  float n = tanhf(inn + r * hn);
  float hv = (1.f - z) * n + z * hp;
  hbuf[(size_t)node * CC + lane] = hv;

  // LayerNorm over the 32 channels (whole wave)
  float sum = hv;
  #pragma unroll
  for (int o = 16; o > 0; o >>= 1) sum += __shfl_xor(sum, o, 32);
  float mu = sum * (1.f / 32.f);
  float dv = hv - mu;
  float s2 = dv * dv;
  #pragma unroll
  for (int o = 16; o > 0; o >>= 1) s2 += __shfl_xor(s2, o, 32);
  float var = s2 * (1.f / 32.f);
  xout[(size_t)node * CC + lane] =
      dv * rsqrtf(var + 1e-5f) * gamma[lane] + beta[lane];
}

// ---------------------------------------------------------------------------
extern "C" void kernel_launch(void* const* d_in, const int* in_sizes, int n_in,
                              void* d_out, int out_size, void* d_ws, size_t ws_size,
                              hipStream_t stream)
{
  const float* x         = (const float*)d_in[0];
  const float* edge_attr = (const float*)d_in[1];
  const int*   eidx      = (const int*)d_in[2];
  const float* w_node    = (const float*)d_in[3];
  const float* w_edge    = (const float*)d_in[4];
  const float* w_att     = (const float*)d_in[5];
  const float* w_scale   = (const float*)d_in[6];
  const float* bias      = (const float*)d_in[7];
  const float* w_ih      = (const float*)d_in[8];
  const float* w_hh      = (const float*)d_in[9];
  const float* b_ih      = (const float*)d_in[10];
  const float* b_hh      = (const float*)d_in[11];
  const float* gamma     = (const float*)d_in[12];
  const float* beta      = (const float*)d_in[13];
  const int* src = eidx;
  const int* dst = eidx + EE;

  float* ws = (float*)d_ws;
  size_t off = 0;
  float* xn    = ws + off; off += (size_t)NN * HH * CC;   // 6.4M
  float* s_i   = ws + off; off += (size_t)NN * HH;
  float* s_j   = ws + off; off += (size_t)NN * HH;
  float* tbuf  = ws + off; off += (size_t)EE * HH;        // 3.2M
  float* alpha = ws + off; off += (size_t)EE * HH;        // 3.2M
  float* amax  = ws + off; off += (size_t)NN * HH;
  float* denom = ws + off; off += (size_t)NN * HH;
  float* aggr  = ws + off; off += (size_t)NN * HH * CC;   // 6.4M
  float* mbuf  = ws + off; off += (size_t)NN * CC;
  float* gi    = ws + off; off += (size_t)NN * 3 * CC;
  float* gh    = ws + off; off += (size_t)NN * 3 * CC;
  float* hbuf  = ws + off; off += (size_t)NN * CC;
  float* xcur  = ws + off; off += (size_t)NN * CC;

  hipMemcpyAsync(xcur, x, sizeof(float) * (size_t)NN * CC,
                 hipMemcpyDeviceToDevice, stream);
  hipMemcpyAsync(hbuf, x, sizeof(float) * (size_t)NN * CC,
                 hipMemcpyDeviceToDevice, stream);

  // Step-invariant edge projection t[e,h]
  t_kernel<<<2048, 256, 0, stream>>>(edge_attr, w_edge, w_att, tbuf);

  const int Mtiles = NN / 16;                 // 3125 exact
  dim3 gemmGrid((Mtiles + 7) / 8);            // 8 waves / block

  for (int s = 0; s < STEPS; ++s) {
    // xn = xcur @ w_node   (K=32, N=128)
    gemm_wmma<CC, HH * CC, 0, 0><<<gemmGrid, 256, 0, stream>>>(
        xcur, w_node, xn, nullptr, Mtiles);

    scores_kernel<<<(NN * HH + 255) / 256, 256, 0, stream>>>(xn, w_att, s_i, s_j);

    init_amax_kernel<<<(NN * HH + 255) / 256, 256, 0, stream>>>(amax, NN * HH);
    hipMemsetAsync(denom, 0, sizeof(float) * (size_t)NN * HH, stream);
    hipMemsetAsync(aggr, 0, sizeof(float) * (size_t)NN * HH * CC, stream);

    alpha_kernel<<<(EE + 255) / 256, 256, 0, stream>>>(src, dst, s_i, s_j,
                                                       tbuf, alpha, amax);
    exp_kernel<<<(EE + 255) / 256, 256, 0, stream>>>(dst, amax, alpha, denom);
    msg_aggr_kernel<<<4096, 128, 0, stream>>>(edge_attr, src, dst, w_edge,
                                              xn, alpha, denom, aggr);

    // m = celu(aggr @ w_scale + bias)   (K=128, N=32)
    gemm_wmma<HH * CC, CC, 0, 1><<<gemmGrid, 256, 0, stream>>>(
        aggr, w_scale, mbuf, bias, Mtiles);

    // gi = m @ w_ih^T ; gh = h @ w_hh^T   (K=32, N=96, B transposed)
    gemm_wmma<CC, 3 * CC, 1, 0><<<gemmGrid, 256, 0, stream>>>(
        mbuf, w_ih, gi, nullptr, Mtiles);
    gemm_wmma<CC, 3 * CC, 1, 0><<<gemmGrid, 256, 0, stream>>>(
        hbuf, w_hh, gh, nullptr, Mtiles);

    gru_ln_kernel<<<(NN * 32 + 255) / 256, 256, 0, stream>>>(
        gi, gh, b_ih, b_hh, hbuf, gamma, beta, xcur);
  }

  hipMemcpyAsync(d_out, xcur, sizeof(float) * (size_t)NN * CC,
                 hipMemcpyDeviceToDevice, stream);
}